// DirectProjecter_57363583205741
// MI455X (gfx1250) — compile-verified
//
#include <hip/hip_runtime.h>
#include <cstdint>

// Problem constants from the reference (fixed by setup_inputs).
#define IMG_H 480
#define IMG_W 640
#define IMG_HW (IMG_H * IMG_W)
#define NBATCH 4

typedef __attribute__((ext_vector_type(4))) float f32x4;
typedef __attribute__((ext_vector_type(2))) unsigned long long u64x2;

// ---------------------------------------------------------------------------
// Pass 0: initialize per-pixel 64-bit keys to {+inf bits, N} ("no winner").
// Two keys per thread via one b128 store.
// ---------------------------------------------------------------------------
__global__ void dp_init_keys(u64x2* __restrict__ keys2, int n2,
                             unsigned long long initv) {
  int t = blockIdx.x * blockDim.x + threadIdx.x;
  if (t < n2) {
    u64x2 v2;
    v2.x = initv;
    v2.y = initv;
    keys2[t] = v2;  // RT store: keys stay resident in L2 for the atomic pass
  }
}

// ---------------------------------------------------------------------------
// Pass 1: scatter-min.  key = {z_bits, point_index}; 64-bit atomicMin gives
// (min depth, then min index among depth ties) lexicographically — identical
// to the reference's two segment_min passes.
// Grid-stride loop, 4 points/iteration via b128 NT loads, prefetch of the
// next iteration's plane lines (global_prefetch_b8) to overlap with the
// in-flight non-returning u64 atomics.
// ---------------------------------------------------------------------------
__global__ void dp_scatter_min(const float* __restrict__ points,
                               unsigned long long* __restrict__ keys,
                               int N, int per_b, int total4) {
  const int stride = gridDim.x * blockDim.x;
  for (int g = blockIdx.x * blockDim.x + threadIdx.x; g < total4; g += stride) {
    int b = g / per_b;
    int i = (g - b * per_b) * 4;

    // Prefetch this thread's next-iteration data while atomics drain.
    int gn = g + stride;
    if (gn < total4) {
      int bn = gn / per_b;
      int in2 = (gn - bn * per_b) * 4;
      const float* nb = points + (size_t)bn * 4 * (size_t)N;
      __builtin_prefetch(nb + 0 * (size_t)N + in2, 0, 1);
      __builtin_prefetch(nb + 1 * (size_t)N + in2, 0, 1);
      __builtin_prefetch(nb + 2 * (size_t)N + in2, 0, 1);
    }

    const float* base = points + (size_t)b * 4 * (size_t)N;  // (4, N) planes
    unsigned long long* kb = keys + (size_t)b * IMG_HW;

    float u[4], v[4], z[4];
    if (((N & 3) == 0) && (i + 3 < N)) {
      // 16B-aligned streaming loads; NT hint keeps L2 free for the key buffer.
      f32x4 uu = __builtin_nontemporal_load((const f32x4*)(base + 0 * (size_t)N + i));
      f32x4 vv = __builtin_nontemporal_load((const f32x4*)(base + 1 * (size_t)N + i));
      f32x4 zz = __builtin_nontemporal_load((const f32x4*)(base + 2 * (size_t)N + i));
      u[0] = uu.x; u[1] = uu.y; u[2] = uu.z; u[3] = uu.w;
      v[0] = vv.x; v[1] = vv.y; v[2] = vv.z; v[3] = vv.w;
      z[0] = zz.x; z[1] = zz.y; z[2] = zz.z; z[3] = zz.w;
    } else {
      #pragma unroll
      for (int k = 0; k < 4; ++k) {
        if (i + k < N) {
          u[k] = base[0 * (size_t)N + i + k];
          v[k] = base[1 * (size_t)N + i + k];
          z[k] = base[2 * (size_t)N + i + k];
        } else {
          u[k] = -1.0f; v[k] = -1.0f; z[k] = -1.0f;  // invalid
        }
      }
    }

    #pragma unroll
    for (int k = 0; k < 4; ++k) {
      int ui = (int)floorf(u[k]);
      int vi = (int)floorf(v[k]);
      float zk = z[k];
      unsigned int zb = __float_as_uint(zk);
      // Reference validity: 0<=u<W, 0<=v<H, z>0, and isfinite(z) to win.
      bool valid = (ui >= 0) & (ui < IMG_W) & (vi >= 0) & (vi < IMG_H) &
                   (zk > 0.0f) & (zb < 0x7F800000u);
      if (valid) {
        int pix = vi * IMG_W + ui;
        unsigned long long key =
            ((unsigned long long)zb << 32) | (unsigned int)(i + k);
        atomicMin(kb + pix, key);  // non-returning -> global_atomic_min_u64
      }
    }
  }
}

// ---------------------------------------------------------------------------
// Pass 2: per-pixel gather of depth / colors / index from the winning key.
// The contiguous key read goes through the CDNA5 async global->LDS path
// (global_load_async_to_lds_b64 + s_wait_asynccnt); random color gathers
// remain direct loads.
// ---------------------------------------------------------------------------
__global__ void __launch_bounds__(256)
dp_gather(const unsigned long long* __restrict__ keys,
          const float* __restrict__ colors,
          float* __restrict__ out, int N) {
  __shared__ unsigned long long skeys[256];
  int t = blockIdx.x * blockDim.x + threadIdx.x;
  bool active = t < NBATCH * IMG_HW;  // grid is an exact fit: always true

  unsigned long long key;
#if defined(__gfx1250__)
  if (active) {
    // Low 32 bits of a generic LDS pointer are the LDS byte offset (ISA 10.2).
    unsigned int ldsoff = (unsigned int)(uintptr_t)&skeys[threadIdx.x];
    unsigned long long ga = (unsigned long long)(uintptr_t)(keys + t);
    asm volatile("global_load_async_to_lds_b64 %0, %1, off"
                 :: "v"(ldsoff), "v"(ga) : "memory");
  }
  asm volatile("s_wait_asynccnt 0x0" ::: "memory");
  key = active ? skeys[threadIdx.x] : 0ull;
#else
  key = active ? keys[t] : 0ull;
#endif
  if (!active) return;

  int b = t / IMG_HW;
  int p = t - b * IMG_HW;

  unsigned int zb  = (unsigned int)(key >> 32);
  unsigned int idx = (unsigned int)key;

  float depth, c0, c1, c2, fidx;
  if (zb >= 0x7F800000u) {  // +inf sentinel: no valid point hit this pixel
    depth = 0.0f; c0 = 0.0f; c1 = 0.0f; c2 = 0.0f;
    fidx = (float)N;
  } else {
    depth = __uint_as_float(zb);
    const float* cb = colors + (size_t)b * 3 * (size_t)N;  // (3, N) planes
    c0 = cb[(size_t)0 * N + idx];
    c1 = cb[(size_t)1 * N + idx];
    c2 = cb[(size_t)2 * N + idx];
    fidx = (float)idx;
  }

  // Output layout: [depth (B,HW)] [img (B,3,HW)] [index (B,HW)]
  const size_t HWs = (size_t)IMG_HW;
  float* depth_out = out;
  float* img_out   = out + (size_t)NBATCH * HWs;
  float* idx_out   = out + (size_t)4 * NBATCH * HWs;

  __builtin_nontemporal_store(depth, depth_out + (size_t)b * HWs + p);
  __builtin_nontemporal_store(c0, img_out + ((size_t)b * 3 + 0) * HWs + p);
  __builtin_nontemporal_store(c1, img_out + ((size_t)b * 3 + 1) * HWs + p);
  __builtin_nontemporal_store(c2, img_out + ((size_t)b * 3 + 2) * HWs + p);
  __builtin_nontemporal_store(fidx, idx_out + (size_t)b * HWs + p);
}

// ---------------------------------------------------------------------------
// Launcher: init -> scatter -> gather, all on `stream` (graph-capture safe).
// ---------------------------------------------------------------------------
extern "C" void kernel_launch(void* const* d_in, const int* in_sizes, int n_in,
                              void* d_out, int out_size, void* d_ws, size_t ws_size,
                              hipStream_t stream) {
  const float* points = (const float*)d_in[0];  // (B, 4, N) f32
  const float* colors = (const float*)d_in[1];  // (B, 3, N) f32
  float* out = (float*)d_out;

  const int N = in_sizes[0] / (4 * NBATCH);

  // Scratch: B*HW u64 keys (9.83 MB) — lives in the 192 MB L2 during pass 1.
  unsigned long long* keys = (unsigned long long*)d_ws;

  const int npix = NBATCH * IMG_HW;
  const int BLK = 256;  // 8 wave32 waves per workgroup

  const unsigned long long initv =
      ((unsigned long long)0x7F800000u << 32) | (unsigned int)N;

  // Pass 0: 2 keys per thread (b128 stores). npix is even.
  const int n2 = npix / 2;
  dp_init_keys<<<(n2 + BLK - 1) / BLK, BLK, 0, stream>>>(
      (u64x2*)keys, n2, initv);

  // Pass 1: grid-stride, ~4 iterations of 4 points per thread.
  const int per_b = (N + 3) / 4;
  const int total4 = NBATCH * per_b;
  int sblocks = 1024;
  dp_scatter_min<<<sblocks, BLK, 0, stream>>>(points, keys, N, per_b, total4);

  // Pass 2: exact-fit grid (npix % 256 == 0 for 480*640*4).
  dp_gather<<<(npix + BLK - 1) / BLK, BLK, 0, stream>>>(keys, colors, out, N);
}